// SumModel_30245159699078
// MI455X (gfx1250) — compile-verified
//
#include <hip/hip_runtime.h>
#include <math.h>

#define BATCH 512
#define SEQ   512
#define EMBED 256
#define NOUT  5

typedef float v2f __attribute__((ext_vector_type(2)));
typedef float v8f __attribute__((ext_vector_type(8)));

// ---------------------------------------------------------------------------
// Kernel 1: ragged-mean embedding pooling.
// grid = BATCH blocks, block = 256 threads (4 token groups x 64 lanes).
// Each group of 64 lanes loads one full 1KB embedding row per iteration as
// float4 (global_load_b128, fully coalesced), with a prefetch 16 tokens ahead
// (global_prefetch_b8). LDS reduces the 4 group-partial vectors.
// ---------------------------------------------------------------------------
__global__ __launch_bounds__(256) void pool_kernel(
    const int* __restrict__ input_x,   // [BATCH, SEQ]
    const int* __restrict__ lengths,   // [BATCH]
    const float* __restrict__ emb,     // [VOCAB, EMBED]
    float* __restrict__ pooled)        // [BATCH, EMBED]
{
    const int b    = blockIdx.x;
    const int t    = threadIdx.x;
    const int g    = t >> 6;     // token subgroup 0..3
    const int lane = t & 63;     // float4 slot within the 256-float row
    const int len  = lengths[b];
    const int* row_idx = input_x + b * SEQ;

    float4 acc = make_float4(0.f, 0.f, 0.f, 0.f);

    for (int s = g; s < len; s += 4) {
        const int tok = row_idx[s];
        // Prefetch a future row of the gather to hide L2 latency.
        const int sp = s + 16;
        if (sp < len) {
            __builtin_prefetch(emb + (size_t)row_idx[sp] * EMBED + lane * 4, 0, 1);
        }
        const float4 v = ((const float4*)(emb + (size_t)tok * EMBED))[lane];
        acc.x += v.x; acc.y += v.y; acc.z += v.z; acc.w += v.w;
    }

    __shared__ float sh[4 * EMBED];
    const int base = g * EMBED + lane * 4;
    sh[base + 0] = acc.x;
    sh[base + 1] = acc.y;
    sh[base + 2] = acc.z;
    sh[base + 3] = acc.w;
    __syncthreads();

    // One thread per embedding dim: sum the 4 group partials, divide by len.
    const int e = t;  // 0..255 == EMBED
    const float inv = 1.0f / (float)len;
    const float s0 = sh[0 * EMBED + e] + sh[1 * EMBED + e];
    const float s1 = sh[2 * EMBED + e] + sh[3 * EMBED + e];
    pooled[(size_t)b * EMBED + e] = (s0 + s1) * inv;
}

// ---------------------------------------------------------------------------
// Kernel 2: head GEMM (pooled @ W.T + b) via V_WMMA_F32_16X16X4_F32, then
// per-row log-softmax + NLL using ds_bpermute shuffles (no LDS needed).
// grid = 32 (one 16-row M-tile per wave32 workgroup), block = 32.
//
// f32 WMMA layouts (ISA 7.12.2):
//   A 16x4 : lane l -> m = l&15, k = 2*(l>>4) + vgpr{0,1}   (contiguous pair)
//   B 4x16 : lane l -> n = l&15, k = 2*(l>>4) + vgpr{0,1}
//   D 16x16: vgpr r, lanes 0-15 -> (M=r,   N=lane),
//                    lanes16-31 -> (M=r+8, N=lane-16)
// ---------------------------------------------------------------------------
__global__ __launch_bounds__(32) void head_kernel(
    const float* __restrict__ pooled,  // [BATCH, EMBED]
    const float* __restrict__ W,       // [NOUT, EMBED]
    const float* __restrict__ bias,    // [NOUT]
    const int*   __restrict__ y,       // [BATCH]
    float* __restrict__ nll)           // [BATCH]
{
    const int tile = blockIdx.x;           // M-tile: rows tile*16 .. tile*16+15
    const int l    = threadIdx.x;          // lane 0..31
    const int half = l >> 4;               // 0 or 1
    const int li   = l & 15;
    const int kb   = half * 2;             // k offset carried by this lane pair

    const bool nvalid = (li < NOUT);

    const float* arow = pooled + ((size_t)(tile * 16 + li)) * EMBED + kb;
    const float* wrow = W + (size_t)(nvalid ? li : 0) * EMBED + kb;  // in-bounds base

    v8f c = {0.f, 0.f, 0.f, 0.f, 0.f, 0.f, 0.f, 0.f};

    for (int k0 = 0; k0 < EMBED; k0 += 4) {
        v2f a;
        a.x = arow[k0];
        a.y = arow[k0 + 1];
        v2f bf;
        const float w0 = wrow[k0];
        const float w1 = wrow[k0 + 1];
        bf.x = nvalid ? w0 : 0.0f;
        bf.y = nvalid ? w1 : 0.0f;
        c = __builtin_amdgcn_wmma_f32_16x16x4_f32(
                /*neg_a=*/false, a, /*neg_b=*/false, bf,
                /*c_mod=*/(short)0, c, /*reuse_a=*/false, /*reuse_b=*/false);
    }

    // Bias: this lane owns column n = li across all 8 rows it holds.
    const float badd = nvalid ? bias[li] : 0.0f;
    const int srcbase = half * 16;  // gather within this lane's half

    for (int r = 0; r < 8; ++r) {
        const float v = c[r] + badd;
        // Row m = r + 8*half of this tile: its 5 logits live in lanes
        // srcbase+0 .. srcbase+4. Broadcast them to every lane in the half.
        const float l0 = __shfl(v, srcbase + 0, 32);
        const float l1 = __shfl(v, srcbase + 1, 32);
        const float l2 = __shfl(v, srcbase + 2, 32);
        const float l3 = __shfl(v, srcbase + 3, 32);
        const float l4 = __shfl(v, srcbase + 4, 32);

        const float mx  = fmaxf(fmaxf(fmaxf(l0, l1), fmaxf(l2, l3)), l4);
        const float se  = __expf(l0 - mx) + __expf(l1 - mx) + __expf(l2 - mx)
                        + __expf(l3 - mx) + __expf(l4 - mx);
        const float lse = mx + __logf(se);

        const int m  = tile * 16 + r + 8 * half;
        const int tg = y[m];
        const float ly = (tg == 0) ? l0 : (tg == 1) ? l1 : (tg == 2) ? l2
                       : (tg == 3) ? l3 : l4;
        if (li == 0) nll[m] = lse - ly;   // -log p[y]
    }
}

// ---------------------------------------------------------------------------
// Kernel 3: deterministic mean of the 512 per-sample NLLs.
// ---------------------------------------------------------------------------
__global__ __launch_bounds__(256) void reduce_kernel(
    const float* __restrict__ nll, float* __restrict__ out)
{
    __shared__ float sh[256];
    const int t = threadIdx.x;
    sh[t] = nll[t] + nll[t + 256];
    __syncthreads();
    for (int off = 128; off > 0; off >>= 1) {
        if (t < off) sh[t] += sh[t + off];
        __syncthreads();
    }
    if (t == 0) out[0] = sh[0] * (1.0f / (float)BATCH);
}

extern "C" void kernel_launch(void* const* d_in, const int* in_sizes, int n_in,
                              void* d_out, int out_size, void* d_ws, size_t ws_size,
                              hipStream_t stream) {
    const int*   input_x = (const int*)d_in[0];    // [BATCH, SEQ]
    const int*   lengths = (const int*)d_in[1];    // [BATCH]
    const int*   input_y = (const int*)d_in[2];    // [BATCH]
    const float* emb     = (const float*)d_in[3];  // [VOCAB, EMBED]
    const float* W       = (const float*)d_in[4];  // [NOUT, EMBED]
    const float* bias    = (const float*)d_in[5];  // [NOUT]
    float* out = (float*)d_out;

    // Workspace layout: pooled [BATCH*EMBED] then nll [BATCH].
    float* pooled = (float*)d_ws;
    float* nll    = pooled + (size_t)BATCH * EMBED;

    pool_kernel<<<BATCH, 256, 0, stream>>>(input_x, lengths, emb, pooled);
    head_kernel<<<BATCH / 16, 32, 0, stream>>>(pooled, W, bias, input_y, nll);
    reduce_kernel<<<1, 256, 0, stream>>>(nll, out);

    (void)in_sizes; (void)n_in; (void)out_size; (void)ws_size;
}